// DetectionCriterion_85143431676239
// MI455X (gfx1250) — compile-verified
//
#include <hip/hip_runtime.h>
#include <stdint.h>

// ---- problem geometry (fixed by the reference) ----
#define NTPL 25
#define BATCH 32
#define HW 16384                  // 128*128
#define SPB (NTPL * HW)           // 409600 elements per batch (class plane)
#define NT (BATCH * SPB)          // 13,107,200 total class elements
#define CLS_STRIDE (125 * HW)     // output batch stride (2,048,000)
#define RM_STRIDE (100 * HW)      // regression_map batch stride
#define KSEL 128u                 // pos_max == neg_max == 128
#define NBINS 65536
#define MINING_THRESH 0.03f
#define REG_WEIGHT 2.0f

typedef __attribute__((ext_vector_type(2))) float v2f;
typedef __attribute__((ext_vector_type(8))) float v8f;

struct Sel { uint32_t cnt, bin1, kprime, thresh; };

__device__ __forceinline__ float softplus_f(float t) {
  // stable softplus: max(t,0) + log1p(exp(-|t|))
  return fmaxf(t, 0.f) + log1pf(__expf(-fabsf(t)));
}
__device__ __forceinline__ float smooth_l1(float d) {
  float ad = fabsf(d);
  return ad < 1.f ? 0.5f * d * d : ad - 0.5f;
}

// Wave32 block reduction using the CDNA5 matrix pipe:
// one V_WMMA_F32_16X16X4_F32 per wave with B = ones reduces 32 lane partials.
__device__ __forceinline__ void block_reduce_atomic(float acc, float* dst) {
  __shared__ float wpart[16];       // 8 waves x 2 halves
  v2f a; a[0] = acc; a[1] = 0.f;    // A[m][0]=acc(lane m), A[m][2]=acc(lane m+16)
  v2f bone; bone[0] = 1.f; bone[1] = 1.f;   // B all-ones: layout-agnostic
  v8f c = {};
  c = __builtin_amdgcn_wmma_f32_16x16x4_f32(false, a, false, bone,
                                            (short)0, c, false, false);
  // D[m][n] = acc_m + acc_{m+16}. Lane0 holds M=0..7 (c[0..7]), lane16 holds M=8..15.
  float s = c[0] + c[1] + c[2] + c[3] + c[4] + c[5] + c[6] + c[7];
  int lane = threadIdx.x & 31;
  int w    = threadIdx.x >> 5;
  if (lane == 0)       wpart[2 * w]     = s;
  else if (lane == 16) wpart[2 * w + 1] = s;
  __syncthreads();
  if (threadIdx.x == 0) {
    float t = 0.f;
#pragma unroll
    for (int i = 0; i < 16; ++i) t += wpart[i];
    atomicAdd(dst, t);
  }
}

// Pass 1: mining + packed cm + top-16-bit noise histogram per (batch, sign).
__global__ void pass1(const float* __restrict__ out, const int* __restrict__ cmap,
                      const float* __restrict__ noise, uint8_t* __restrict__ cm8,
                      uint32_t* __restrict__ hist1) {
  const int nq = NT / 4;
  const int stride = gridDim.x * blockDim.x;
  for (int q = blockIdx.x * blockDim.x + threadIdx.x; q < nq; q += stride) {
    int base = q * 4;
    int b = base / SPB;
    int r = base - b * SPB;
    __builtin_prefetch(cmap + base + 4 * stride, 0, 0);
    __builtin_prefetch(noise + base + 4 * stride, 0, 0);
    int4   cmi = *(const int4*)(cmap + base);
    float4 nz  = *(const float4*)(noise + base);
    float4 cl  = *(const float4*)(out + b * CLS_STRIDE + r);
    const int   cv[4] = {cmi.x, cmi.y, cmi.z, cmi.w};
    const float cf[4] = {cl.x, cl.y, cl.z, cl.w};
    const float nf[4] = {nz.x, nz.y, nz.z, nz.w};
    uint32_t packed = 0;
#pragma unroll
    for (int j = 0; j < 4; ++j) {
      int cm = cv[j] - 1;                       // -1, 0, +1
      if (cm != 0) {
        float sp = softplus_f(-cf[j] * (float)cm);
        if (sp < MINING_THRESH) cm = 0;         // mining: drop easy
      }
      if (cm != 0) {
        int s = (cm == 1) ? 0 : 1;
        uint32_t bits = __float_as_uint(nf[j]); // monotonic for noise in [0,1)
        atomicAdd(&hist1[(uint32_t)(b * 2 + s) * NBINS + (bits >> 16)], 1u);
      }
      packed |= ((uint32_t)(uint8_t)(int8_t)cm) << (8 * j);
    }
    *(uint32_t*)(cm8 + base) = packed;
  }
}

// Find the bin containing the K-th smallest; one block per (batch,sign).
__global__ void findbin1(const uint32_t* __restrict__ hist, Sel* __restrict__ sel) {
  const uint32_t* h = hist + (size_t)blockIdx.x * NBINS;
  __shared__ uint32_t csum[256];
  uint32_t s = 0;
  int cbase = threadIdx.x * 256;
  for (int i = 0; i < 256; ++i) s += h[cbase + i];
  csum[threadIdx.x] = s;
  __syncthreads();
  if (threadIdx.x == 0) {
    uint32_t total = 0;
    for (int i = 0; i < 256; ++i) total += csum[i];
    Sel o; o.cnt = total;
    if (total <= KSEL) {
      o.bin1 = 0; o.kprime = 0; o.thresh = 0xFFFFFFFFu;   // keep all flagged
    } else {
      uint32_t cum = 0; int c = 0;
      while (c < 255 && cum + csum[c] < KSEL) { cum += csum[c]; ++c; }
      uint32_t cum2 = cum; int bin = c * 256; uint32_t kp = 1;
      for (int i = 0; i < 256; ++i) {
        uint32_t hv = h[c * 256 + i];
        if (cum2 + hv >= KSEL) { bin = c * 256 + i; kp = KSEL - cum2; break; }
        cum2 += hv;
      }
      o.bin1 = (uint32_t)bin; o.kprime = kp; o.thresh = 0u;  // pending refine
    }
    sel[blockIdx.x] = o;
  }
}

// Pass 2: low-16-bit refinement histogram restricted to the boundary bin.
__global__ void pass2(const uint8_t* __restrict__ cm8, const float* __restrict__ noise,
                      const Sel* __restrict__ sel, uint32_t* __restrict__ hist2) {
  const int nq = NT / 4;
  const int stride = gridDim.x * blockDim.x;
  for (int q = blockIdx.x * blockDim.x + threadIdx.x; q < nq; q += stride) {
    int base = q * 4;
    uint32_t packed = *(const uint32_t*)(cm8 + base);
    if (packed == 0u) continue;
    int b = base / SPB;
    float4 nz = *(const float4*)(noise + base);
    const float nf[4] = {nz.x, nz.y, nz.z, nz.w};
    Sel sp = sel[b * 2 + 0];
    Sel sn = sel[b * 2 + 1];
#pragma unroll
    for (int j = 0; j < 4; ++j) {
      int cm = (int)(int8_t)(packed >> (8 * j));
      if (cm == 0) continue;
      int s = (cm == 1) ? 0 : 1;
      const Sel& qs = (s == 0) ? sp : sn;
      if (qs.thresh == 0xFFFFFFFFu) continue;  // keep-all: no refine needed
      uint32_t bits = __float_as_uint(nf[j]);
      if ((bits >> 16) == qs.bin1)
        atomicAdd(&hist2[(uint32_t)(b * 2 + s) * NBINS + (bits & 0xFFFFu)], 1u);
    }
  }
}

// Resolve exact 32-bit threshold (bits of the K-th smallest flagged noise).
__global__ void findbin2(const uint32_t* __restrict__ hist, Sel* __restrict__ sel) {
  Sel sv = sel[blockIdx.x];
  if (sv.thresh == 0xFFFFFFFFu) return;
  const uint32_t* h = hist + (size_t)blockIdx.x * NBINS;
  __shared__ uint32_t csum[256];
  uint32_t s = 0;
  int cbase = threadIdx.x * 256;
  for (int i = 0; i < 256; ++i) s += h[cbase + i];
  csum[threadIdx.x] = s;
  __syncthreads();
  if (threadIdx.x == 0) {
    uint32_t K = sv.kprime;
    uint32_t cum = 0; int c = 0;
    while (c < 255 && cum + csum[c] < K) { cum += csum[c]; ++c; }
    uint32_t cum2 = cum; int bin = c * 256 + 255;
    for (int i = 0; i < 256; ++i) {
      uint32_t hv = h[c * 256 + i];
      if (cum2 + hv >= K) { bin = c * 256 + i; break; }
      cum2 += hv;
    }
    sel[blockIdx.x].thresh = (sv.bin1 << 16) | (uint32_t)bin;
  }
}

// Final: stream cm8+noise; gather cls / reg / regression_map only at kept spots.
__global__ void finalk(const float* __restrict__ out, const float* __restrict__ rmap,
                       const float* __restrict__ noise, const uint8_t* __restrict__ cm8,
                       const Sel* __restrict__ sel, float* __restrict__ dst) {
  float acc = 0.f;
  const int nq = NT / 4;
  const int stride = gridDim.x * blockDim.x;
  for (int q = blockIdx.x * blockDim.x + threadIdx.x; q < nq; q += stride) {
    int base = q * 4;
    uint32_t packed = *(const uint32_t*)(cm8 + base);
    if (packed == 0u) continue;
    int b  = base / SPB;
    int r0 = base - b * SPB;
    float4 nz = *(const float4*)(noise + base);
    const float nf[4] = {nz.x, nz.y, nz.z, nz.w};
    uint32_t thp = sel[b * 2 + 0].thresh;
    uint32_t thn = sel[b * 2 + 1].thresh;
#pragma unroll
    for (int j = 0; j < 4; ++j) {
      int cm = (int)(int8_t)(packed >> (8 * j));
      if (cm == 0) continue;
      uint32_t bits = __float_as_uint(nf[j]);
      uint32_t th = (cm == 1) ? thp : thn;
      if (bits > th) continue;                   // not sampled
      int r = r0 + j;
      float cls = out[b * CLS_STRIDE + r];       // sparse gather
      acc += softplus_f(-cls * (float)cm);
      if (cm == 1) {
        int t  = r / HW;
        int hw = r - t * HW;
        const float* ob = out  + b * CLS_STRIDE + (NTPL + t) * HW + hw;
        const float* mb = rmap + b * RM_STRIDE  + t * HW + hw;
#pragma unroll
        for (int c2 = 0; c2 < 4; ++c2) {
          float d = ob[c2 * NTPL * HW] - mb[c2 * NTPL * HW];
          acc += REG_WEIGHT * smooth_l1(d);
        }
      }
    }
  }
  block_reduce_atomic(acc, dst);
}

extern "C" void kernel_launch(void* const* d_in, const int* in_sizes, int n_in,
                              void* d_out, int out_size, void* d_ws, size_t ws_size,
                              hipStream_t stream) {
  const float* output         = (const float*)d_in[0];
  const int*   class_map      = (const int*)d_in[1];
  const float* regression_map = (const float*)d_in[2];
  const float* noise          = (const float*)d_in[3];
  float* dst = (float*)d_out;

  // workspace layout: cm8 | hist1 | hist2 | sel
  uint8_t* ws = (uint8_t*)d_ws;
  uint8_t*  cm8   = ws;
  size_t off1 = (size_t)NT;                         // 13,107,200 B (16B aligned)
  uint32_t* hist1 = (uint32_t*)(ws + off1);
  size_t off2 = off1 + (size_t)64 * NBINS * 4;
  uint32_t* hist2 = (uint32_t*)(ws + off2);
  size_t off3 = off2 + (size_t)64 * NBINS * 4;
  Sel* sel = (Sel*)(ws + off3);

  hipMemsetAsync(ws + off1, 0, (size_t)64 * NBINS * 4 * 2 + 64 * sizeof(Sel), stream);
  hipMemsetAsync(d_out, 0, sizeof(float), stream);

  dim3 blk(256);     // 8 waves (wave32)
  dim3 grd(2048);
  pass1<<<grd, blk, 0, stream>>>(output, class_map, noise, cm8, hist1);
  findbin1<<<dim3(64), blk, 0, stream>>>(hist1, sel);
  pass2<<<grd, blk, 0, stream>>>(cm8, noise, sel, hist2);
  findbin2<<<dim3(64), blk, 0, stream>>>(hist2, sel);
  finalk<<<grd, blk, 0, stream>>>(output, regression_map, noise, cm8, sel, dst);
}